// EnhancedQuanvolution_65481071399704
// MI455X (gfx1250) — compile-verified
//
#include <hip/hip_runtime.h>

typedef __attribute__((ext_vector_type(2))) float v2f;
typedef __attribute__((ext_vector_type(8))) float v8f;

// One wave (32 lanes) produces logits for 16 images via V_WMMA_F32_16X16X4_F32.
// K-chunk of 4 == one 2x2 patch's cumprod(cos) features.
// A fragment (16x4 f32): lanes 0-15 -> K=0,1 in v0,v1 ; lanes 16-31 -> K=2,3.
// B fragment (4x16 f32): lanes 0-15 -> K=0 (v0), K=1 (v1); lanes 16-31 -> K=2,3; N=lane&15.
// C/D (16x16 f32): acc[j] = D[M=j+8*(lane>=16)][N=lane&15].

__global__ __launch_bounds__(256)
void quanv_wmma_kernel(const float* __restrict__ x,
                       const float* __restrict__ W,
                       const float* __restrict__ bias,
                       float* __restrict__ out)
{
    // Padded W^T staged in LDS, pre-swizzled so each lane's B-fragment pair
    // (v0,v1) is one contiguous 8-byte ds_load: lds_w[chunk*64 + lane*2 + {0,1}]
    __shared__ float lds_w[196 * 64];   // 50176 B

    const int tid = threadIdx.x;
    for (int idx = tid; idx < 196 * 64; idx += 256) {
        int c    = idx >> 6;          // chunk 0..195
        int lane = (idx >> 1) & 31;
        int half = idx & 1;
        int k    = 4 * c + 2 * (lane >> 4) + half;   // 0..783
        int n    = lane & 15;
        lds_w[idx] = (n < 10) ? W[n * 784 + k] : 0.0f;
    }
    __syncthreads();

    const int lane = tid & 31;
    const int wave = tid >> 5;
    const int n    = lane & 15;
    const bool hi  = lane >= 16;
    const int m0   = blockIdx.x * 128 + wave * 16;   // first image of this wave's tile
    const float* __restrict__ px = x + (size_t)(m0 + n) * 784;

    v8f acc = {};

    for (int r = 0; r < 14; ++r) {
        const float* row0 = px + (2 * r) * 28;
        const float* row1 = row0 + 28;
        #pragma unroll
        for (int c2 = 0; c2 < 7; ++c2) {
            // 16-byte loads cover two adjacent patches (cols 4c2 .. 4c2+3)
            float4 a0 = *(const float4*)(row0 + 4 * c2);
            float4 a1 = *(const float4*)(row1 + 4 * c2);
            int chunk = r * 14 + 2 * c2;

            {   // patch (r, 2*c2): pixels a0.x a0.y / a1.x a1.y
                float e0 = __cosf(a0.x);
                float e1 = e0 * __cosf(a0.y);
                float e2 = e1 * __cosf(a1.x);
                float e3 = e2 * __cosf(a1.y);
                v2f a; a.x = hi ? e2 : e0; a.y = hi ? e3 : e1;
                v2f bf = *(const v2f*)(lds_w + chunk * 64 + lane * 2);
                acc = __builtin_amdgcn_wmma_f32_16x16x4_f32(
                          false, a, false, bf, (short)0, acc, false, false);
            }
            {   // patch (r, 2*c2+1): pixels a0.z a0.w / a1.z a1.w
                float e0 = __cosf(a0.z);
                float e1 = e0 * __cosf(a0.w);
                float e2 = e1 * __cosf(a1.z);
                float e3 = e2 * __cosf(a1.w);
                v2f a; a.x = hi ? e2 : e0; a.y = hi ? e3 : e1;
                v2f bf = *(const v2f*)(lds_w + (chunk + 1) * 64 + lane * 2);
                acc = __builtin_amdgcn_wmma_f32_16x16x4_f32(
                          false, a, false, bf, (short)0, acc, false, false);
            }
        }
    }

    // Bias + log-softmax across N (16-lane groups hold one logit row per acc[j]).
    const float bv = (n < 10) ? bias[n] : 0.0f;
    const float NEGBIG = -1e30f;
    #pragma unroll
    for (int j = 0; j < 8; ++j) {
        float v = (n < 10) ? (acc[j] + bv) : NEGBIG;   // pad lanes masked out
        float m = v;
        #pragma unroll
        for (int off = 1; off < 16; off <<= 1)
            m = fmaxf(m, __shfl_xor(m, off, 16));
        float e = __expf(v - m);                        // pad lanes -> exp(-huge)=0
        float s = e;
        #pragma unroll
        for (int off = 1; off < 16; off <<= 1)
            s += __shfl_xor(s, off, 16);
        float res = v - m - __logf(s);
        int b = m0 + j + (hi ? 8 : 0);
        if (n < 10)
            out[b * 10 + n] = res;
    }
}

extern "C" void kernel_launch(void* const* d_in, const int* in_sizes, int n_in,
                              void* d_out, int out_size, void* d_ws, size_t ws_size,
                              hipStream_t stream) {
    const float* x = (const float*)d_in[0];   // (65536,1,28,28) f32
    const float* W = (const float*)d_in[1];   // (10,784) f32
    const float* b = (const float*)d_in[2];   // (10,) f32
    float* out = (float*)d_out;               // (65536,10) f32

    const int batch  = in_sizes[0] / 784;     // 65536
    const int blocks = batch / 128;           // 128 images per 256-thread block
    quanv_wmma_kernel<<<blocks, 256, 0, stream>>>(x, W, b, out);
}